// Qwen3Attention_85040352461107
// MI455X (gfx1250) — compile-verified
//
#include <hip/hip_runtime.h>
#include <hip/hip_bf16.h>
#include <math.h>

// ---------------------------------------------------------------------------
// Qwen3 decode attention layer for MI455X (gfx1250), fp32 end-to-end.
// Memory-bound workload (~300 MB traffic @ 23.3 TB/s); fp32 WMMA keeps math
// exact while exercising the CDNA5 matrix path. KV staging uses the CDNA5
// async memory->LDS path (GLOBAL_LOAD_ASYNC_TO_LDS_B128, ASYNCcnt).
// ---------------------------------------------------------------------------

#define BATCH      32
#define HIDDEN     2048
#define NUM_HEADS  16
#define NUM_KVH    8
#define HEAD_DIM   128
#define BLOCK_SZ   16
#define MAX_BLOCKS 128

#define USE_ASYNC_LDS 1

typedef float v2f __attribute__((ext_vector_type(2)));
typedef float v8f __attribute__((ext_vector_type(8)));

// ---------------------------------------------------------------------------
// Generic fp32 WMMA GEMM: C[Mtiles*16 x N] = A[.. x K] * B[K x N], row-major.
// One wave per 16x16 C tile, K stepped by 4 via V_WMMA_F32_16X16X4_F32.
// A fragment: lane (m = lane&15, g = lane>>4) holds A[m][k0+2g], A[m][k0+2g+1].
// B fragment: same lane holds B[k0+2g][n], B[k0+2g+1][n].
// C fragment: c[r] is C[row0 + r + 8g][col0 + (lane&15)].
// ---------------------------------------------------------------------------
__global__ __launch_bounds__(32) void gemm16_wmma_f32(
    const float* __restrict__ A, const float* __restrict__ B,
    float* __restrict__ C, int K, int N)
{
    const int lane = threadIdx.x & 31;
    const int mn   = lane & 15;
    const int grp  = lane >> 4;
    const int row0 = blockIdx.y << 4;
    const int col0 = blockIdx.x << 4;

    const float* ap = A + (size_t)(row0 + mn) * K + 2 * grp;
    const float* bp = B + (size_t)(2 * grp) * N + col0 + mn;

    v8f c = {0.f, 0.f, 0.f, 0.f, 0.f, 0.f, 0.f, 0.f};

#pragma unroll 8
    for (int k0 = 0; k0 < K; k0 += 4) {
        float2 av = *(const float2*)ap;
        ap += 4;
        v2f a; a.x = av.x; a.y = av.y;
        v2f b; b.x = bp[0]; b.y = bp[N];
        bp += 4 * N;
        // 8 args: (neg_a, A, neg_b, B, c_mod, C, reuse_a, reuse_b)
        c = __builtin_amdgcn_wmma_f32_16x16x4_f32(
            false, a, false, b, (short)0, c, false, false);
    }

#pragma unroll
    for (int r = 0; r < 8; ++r)
        C[(size_t)(row0 + r + 8 * grp) * N + col0 + mn] = c[r];
}

// ---------------------------------------------------------------------------
// Per-head RMSNorm + RoPE for q and k; pass-through copy for v.
// grid = (BATCH, NUM_HEADS + 2*NUM_KVH), block = HEAD_DIM threads.
//   rows [0,16):  q head  -> ws_qf
//   rows [16,24): k head  -> d_out k-section (also the returned k)
//   rows [24,32): v head  -> d_out v-section (raw copy)
// ---------------------------------------------------------------------------
__global__ __launch_bounds__(HEAD_DIM) void normrope_kernel(
    const float* __restrict__ ws_q, const float* __restrict__ ws_k,
    const float* __restrict__ ws_v,
    const float* __restrict__ cosb, const float* __restrict__ sinb,
    const float* __restrict__ qw, const float* __restrict__ kw,
    float* __restrict__ qf, float* __restrict__ kout, float* __restrict__ vout)
{
    const int b   = blockIdx.x;
    const int r   = blockIdx.y;
    const int tid = threadIdx.x;

    if (r >= NUM_HEADS + NUM_KVH) {              // v: raw copy
        const int g = r - (NUM_HEADS + NUM_KVH);
        vout[(b * NUM_KVH + g) * HEAD_DIM + tid] =
            ws_v[(b * NUM_KVH + g) * HEAD_DIM + tid];
        return;
    }

    const float* src; const float* wvec; float* dst;
    if (r < NUM_HEADS) {
        src  = ws_q + (size_t)(b * NUM_HEADS + r) * HEAD_DIM;
        wvec = qw;
        dst  = qf + (size_t)(b * NUM_HEADS + r) * HEAD_DIM;
    } else {
        const int g = r - NUM_HEADS;
        src  = ws_k + (size_t)(b * NUM_KVH + g) * HEAD_DIM;
        wvec = kw;
        dst  = kout + (size_t)(b * NUM_KVH + g) * HEAD_DIM;
    }

    __shared__ float red[HEAD_DIM];
    __shared__ float xn[HEAD_DIM];

    const float x = src[tid];
    red[tid] = x * x;
    __syncthreads();
#pragma unroll
    for (int s = HEAD_DIM / 2; s > 0; s >>= 1) {
        if (tid < s) red[tid] += red[tid + s];
        __syncthreads();
    }
    const float inv = rsqrtf(red[0] * (1.0f / HEAD_DIM) + 1e-6f);
    const float nx  = x * inv * wvec[tid];
    xn[tid] = nx;
    __syncthreads();

    const float rot = (tid < HEAD_DIM / 2) ? -xn[tid + HEAD_DIM / 2]
                                           :  xn[tid - HEAD_DIM / 2];
    dst[tid] = nx * cosb[b * HEAD_DIM + tid] + rot * sinb[b * HEAD_DIM + tid];
}

// ---------------------------------------------------------------------------
// Paged decode attention, online softmax. One 128-thread block per (b, qh).
// n_rep = 2 query heads per kv head. KV tiles staged into LDS via CDNA5
// GLOBAL_LOAD_ASYNC_TO_LDS_B128 (ASYNCcnt), drained with s_wait_asynccnt;
// scores reduced with wave32 shfl_xor; next block prefetched.
// ---------------------------------------------------------------------------
__global__ __launch_bounds__(HEAD_DIM) void paged_attn_kernel(
    const float* __restrict__ qf,
    const float* __restrict__ k_cache, const float* __restrict__ v_cache,
    const int* __restrict__ block_tables, const int* __restrict__ context_lens,
    float* __restrict__ attn_out)
{
    const int b    = blockIdx.x;
    const int qh   = blockIdx.y;
    const int g    = qh >> 1;                 // kv head (n_rep = 2)
    const int tid  = threadIdx.x;
    const int lane = tid & 31;
    const int w    = tid >> 5;

    __shared__ __align__(16) float kt[BLOCK_SZ * HEAD_DIM];
    __shared__ __align__(16) float vt[BLOCK_SZ * HEAD_DIM];
    __shared__ float sc[BLOCK_SZ];
    __shared__ float part[4][BLOCK_SZ];

    const float scale = 0.08838834764831845f; // 1/sqrt(128)
    const float q     = qf[(size_t)(b * NUM_HEADS + qh) * HEAD_DIM + tid] * scale;

    const int ctx     = context_lens[b];
    const int nblocks = (ctx + BLOCK_SZ - 1) >> 4;
    const float NEG_INF = -__builtin_inff();

    float m = NEG_INF, l = 0.f, acc = 0.f;

    for (int jb = 0; jb < nblocks; ++jb) {
        const int blk = block_tables[b * MAX_BLOCKS + jb];
        // element (blk, t, g, d) at ((blk*16 + t)*8 + g)*128 + d
        const float* kbase = k_cache + (size_t)blk * (BLOCK_SZ * NUM_KVH * HEAD_DIM)
                                     + (size_t)g * HEAD_DIM;
        const float* vbase = v_cache + (size_t)blk * (BLOCK_SZ * NUM_KVH * HEAD_DIM)
                                     + (size_t)g * HEAD_DIM;

        if (jb + 1 < nblocks) {                 // gfx1250 global_prefetch_b8
            const int nblk = block_tables[b * MAX_BLOCKS + jb + 1];
            const float* pk = k_cache + (size_t)nblk * (BLOCK_SZ * NUM_KVH * HEAD_DIM)
                                      + (size_t)g * HEAD_DIM;
            const float* pv = v_cache + (size_t)nblk * (BLOCK_SZ * NUM_KVH * HEAD_DIM)
                                      + (size_t)g * HEAD_DIM;
            __builtin_prefetch(pk, 0, 1);
            __builtin_prefetch(pv, 0, 1);
        }

        // Stage 16x128 K and V tiles into LDS, 16 bytes per lane per issue.
#pragma unroll
        for (int i = tid; i < BLOCK_SZ * HEAD_DIM / 4; i += HEAD_DIM) {
            const int t  = i >> 5;              // 32 float4 per token row
            const int d4 = (i & 31) << 2;
            const size_t off = (size_t)t * (NUM_KVH * HEAD_DIM) + d4;
#if USE_ASYNC_LDS
            // CDNA5 async memory->LDS copy; no VGPR transit. VDST carries the
            // LDS byte address (low 32 bits of the shared pointer), VADDR the
            // 64-bit global address, SADDR=off.
            const unsigned lds_k = (unsigned)(size_t)&kt[t * HEAD_DIM + d4];
            const unsigned lds_v = (unsigned)(size_t)&vt[t * HEAD_DIM + d4];
            asm volatile("global_load_async_to_lds_b128 %0, %1, off"
                         :: "v"(lds_k), "v"(kbase + off) : "memory");
            asm volatile("global_load_async_to_lds_b128 %0, %1, off"
                         :: "v"(lds_v), "v"(vbase + off) : "memory");
#else
            *(float4*)(&kt[t * HEAD_DIM + d4]) = *(const float4*)(kbase + off);
            *(float4*)(&vt[t * HEAD_DIM + d4]) = *(const float4*)(vbase + off);
#endif
        }
#if USE_ASYNC_LDS
        // Drain this wave's ASYNCcnt, then barrier so every wave's tiles are
        // resident before anyone reads LDS.
#if __has_builtin(__builtin_amdgcn_s_wait_asynccnt)
        __builtin_amdgcn_s_wait_asynccnt(0);
#else
        asm volatile("s_wait_asynccnt 0x0" ::: "memory");
#endif
#endif
        __syncthreads();

        // Partial scores: thread tid contributes q[tid]*K[t][tid] for all t.
        float ps[BLOCK_SZ];
#pragma unroll
        for (int t = 0; t < BLOCK_SZ; ++t)
            ps[t] = q * kt[t * HEAD_DIM + tid];

        // wave32 reduction, then cross-wave via LDS
#pragma unroll
        for (int t = 0; t < BLOCK_SZ; ++t) {
            float v = ps[t];
            for (int off = 16; off >= 1; off >>= 1)
                v += __shfl_xor(v, off, 32);
            if (lane == 0) part[w][t] = v;
        }
        __syncthreads();
        if (tid < BLOCK_SZ) {
            const float s = part[0][tid] + part[1][tid] + part[2][tid] + part[3][tid];
            const int gt  = jb * BLOCK_SZ + tid;
            sc[tid] = (gt < ctx) ? s : NEG_INF;
        }
        __syncthreads();

        // Online softmax update (computed redundantly by every thread).
        float mt = m;
#pragma unroll
        for (int t = 0; t < BLOCK_SZ; ++t) mt = fmaxf(mt, sc[t]);
        const float corr = __expf(m - mt);      // m=-inf on first tile -> 0
        float psum = 0.f, a2 = 0.f;
#pragma unroll
        for (int t = 0; t < BLOCK_SZ; ++t) {
            const float p = __expf(sc[t] - mt); // masked -> exp(-inf) = 0
            psum += p;
            a2   += p * vt[t * HEAD_DIM + tid];
        }
        l   = l * corr + psum;
        acc = acc * corr + a2;
        m   = mt;
        __syncthreads();                        // before LDS reuse
    }

    attn_out[(size_t)(b * NUM_HEADS + qh) * HEAD_DIM + tid] = acc / l;
}

// ---------------------------------------------------------------------------
// Launch: qkv GEMMs -> norm/rope -> paged attention -> output GEMM.
// ws layout (floats): q_raw 65536 | k_raw 32768 | v_raw 32768 | q_final 65536
//                     | attn_out 65536  (total 1 MB)
// ---------------------------------------------------------------------------
extern "C" void kernel_launch(void* const* d_in, const int* in_sizes, int n_in,
                              void* d_out, int out_size, void* d_ws, size_t ws_size,
                              hipStream_t stream)
{
    const float* hidden  = (const float*)d_in[0];
    const float* cosb    = (const float*)d_in[1];
    const float* sinb    = (const float*)d_in[2];
    const float* k_cache = (const float*)d_in[3];
    const float* v_cache = (const float*)d_in[4];
    const float* wq      = (const float*)d_in[5];
    const float* wk      = (const float*)d_in[6];
    const float* wv      = (const float*)d_in[7];
    const float* wo      = (const float*)d_in[8];
    const float* qnw     = (const float*)d_in[9];
    const float* knw     = (const float*)d_in[10];
    const int*   btab    = (const int*)d_in[11];
    const int*   clen    = (const int*)d_in[12];

    float* out  = (float*)d_out;                       // 32 x 2048
    float* kout = out + BATCH * HIDDEN;                // 32 x 8 x 128
    float* vout = kout + BATCH * NUM_KVH * HEAD_DIM;   // 32 x 8 x 128

    float* ws      = (float*)d_ws;
    float* ws_q    = ws;                               // 32 x 2048
    float* ws_k    = ws_q + BATCH * NUM_HEADS * HEAD_DIM;   // 32 x 1024
    float* ws_v    = ws_k + BATCH * NUM_KVH * HEAD_DIM;     // 32 x 1024
    float* ws_qf   = ws_v + BATCH * NUM_KVH * HEAD_DIM;     // 32 x 2048
    float* ws_attn = ws_qf + BATCH * NUM_HEADS * HEAD_DIM;  // 32 x 2048

    // QKV projections (fp32 WMMA)
    gemm16_wmma_f32<<<dim3(NUM_HEADS * HEAD_DIM / 16, BATCH / 16), 32, 0, stream>>>(
        hidden, wq, ws_q, HIDDEN, NUM_HEADS * HEAD_DIM);
    gemm16_wmma_f32<<<dim3(NUM_KVH * HEAD_DIM / 16, BATCH / 16), 32, 0, stream>>>(
        hidden, wk, ws_k, HIDDEN, NUM_KVH * HEAD_DIM);
    gemm16_wmma_f32<<<dim3(NUM_KVH * HEAD_DIM / 16, BATCH / 16), 32, 0, stream>>>(
        hidden, wv, ws_v, HIDDEN, NUM_KVH * HEAD_DIM);

    // RMSNorm + RoPE (q, k) and v pass-through
    normrope_kernel<<<dim3(BATCH, NUM_HEADS + 2 * NUM_KVH), HEAD_DIM, 0, stream>>>(
        ws_q, ws_k, ws_v, cosb, sinb, qnw, knw, ws_qf, kout, vout);

    // Paged attention
    paged_attn_kernel<<<dim3(BATCH, NUM_HEADS), HEAD_DIM, 0, stream>>>(
        ws_qf, k_cache, v_cache, btab, clen, ws_attn);

    // Output projection (fp32 WMMA)
    gemm16_wmma_f32<<<dim3(HIDDEN / 16, BATCH / 16), 32, 0, stream>>>(
        ws_attn, wo, out, HIDDEN, HIDDEN);
}